// ProbAttention_15908558864588
// MI455X (gfx1250) — compile-verified
//
#include <hip/hip_runtime.h>

// ---------------------------------------------------------------------------
// ProbAttention (Informer-style, degenerate-dense case) for MI455X / gfx1250.
// B=16, L=256, H=8, D=256, FACTOR=48  ->  U_part = u = 256 (fully dense).
//
//   S  = Q K^T            (per b,h 256x256x256, fp32 WMMA 16x16x4)
//   M  = max_j S[q,idx] - mean_j S[q,idx]   (gathered samples)
//   M_top = stable descending argsort of M  (rank-by-count, matches top_k)
//   W^T = V Q^T           (stored transposed so phase-3 B-loads are b64)
//   out = mean_l V  +  (1/16) * S[M_top,:] @ W     (written as (B,L,H,D))
//
// Workspace layout in d_ws (needs ~64.3 MB):
//   [0)              S   : 128 * 256*256 f32  (32 MB)
//   [32MB)           W^T : 128 * 256*256 f32  (32 MB)
//   [64MB)           M_top: 128 * 256 int32   (128 KB)
//   [64MB+128KB)     vmean: 128 * 256 f32     (128 KB)
// ---------------------------------------------------------------------------

typedef __attribute__((ext_vector_type(2))) float v2f;
typedef __attribute__((ext_vector_type(8))) float v8f;

#define B_BATCH 16
#define L_SEQ   256
#define H_HEADS 8
#define D_DIM   256
#define BH      (B_BATCH * H_HEADS)      /* 128  */
#define ROWSTR  (H_HEADS * D_DIM)        /* 2048 floats between consecutive l */
#define MAT_ELEMS (L_SEQ * L_SEQ)        /* 65536 */
#define SCALE   0.0625f                  /* 1/sqrt(256) */

static __device__ __forceinline__ v8f wmma_f32_k4(v2f a, v2f b, v8f c) {
  // V_WMMA_F32_16X16X4_F32 : D = A(16x4) * B(4x16) + C, fp32 in/out
  return __builtin_amdgcn_wmma_f32_16x16x4_f32(
      /*neg_a=*/false, a, /*neg_b=*/false, b,
      /*c_mod=*/(short)0, c, /*reuse_a=*/false, /*reuse_b=*/false);
}

// ---------------------------------------------------------------------------
// Phase 1: z even -> S = Q K^T   (A = Q rows, B = K rows, contraction on D)
//          z odd  -> W^T = V Q^T (A = V rows, B = Q rows)
// Both stored row-major 256x256 per (b,h). Wave computes a 2x4 tile block.
// ---------------------------------------------------------------------------
__global__ __launch_bounds__(256) void pa_phase1(
    const float* __restrict__ Q, const float* __restrict__ K,
    const float* __restrict__ V, float* __restrict__ Sout,
    float* __restrict__ WTout)
{
  const int z   = blockIdx.y;          // 0 .. 2*BH-1
  const int bh  = z >> 1;
  const int isW = z & 1;
  const int b   = bh >> 3;             // bh / H
  const int h   = bh & (H_HEADS - 1);

  const size_t headOff = (size_t)b * L_SEQ * ROWSTR + (size_t)h * D_DIM;
  const float* Abase = (isW ? V : Q) + headOff;
  const float* Bbase = (isW ? Q : K) + headOff;
  float* Cout = (isW ? WTout : Sout) + (size_t)bh * MAT_ELEMS;

  const int wave = threadIdx.x >> 5;
  const int lane = threadIdx.x & 31;
  const int lm   = lane & 15;
  const int kk   = (lane >> 4) << 1;   // 0 or 2 : K-pair base within k-step

  const int tr0 = blockIdx.x * 4 + (wave >> 2) * 2;  // tile rows tr0, tr0+1
  const int tc0 = (wave & 3) * 4;                    // tile cols tc0..tc0+3

  // A fragment rows (row L&15, floats k0+kk .. k0+kk+1) -> one b64 per step
  const float* ap0 = Abase + (size_t)(tr0 * 16 + lm)       * ROWSTR + kk;
  const float* ap1 = Abase + (size_t)((tr0 + 1) * 16 + lm) * ROWSTR + kk;
  const float* bp0 = Bbase + (size_t)((tc0 + 0) * 16 + lm) * ROWSTR + kk;
  const float* bp1 = Bbase + (size_t)((tc0 + 1) * 16 + lm) * ROWSTR + kk;
  const float* bp2 = Bbase + (size_t)((tc0 + 2) * 16 + lm) * ROWSTR + kk;
  const float* bp3 = Bbase + (size_t)((tc0 + 3) * 16 + lm) * ROWSTR + kk;

  const v8f zero = {0.f, 0.f, 0.f, 0.f, 0.f, 0.f, 0.f, 0.f};
  v8f acc[2][4];
  for (int i = 0; i < 2; ++i)
    for (int j = 0; j < 4; ++j) acc[i][j] = zero;

#pragma unroll 8
  for (int k0 = 0; k0 < D_DIM; k0 += 4) {
    v2f a0 = *(const v2f*)(ap0 + k0);
    v2f a1 = *(const v2f*)(ap1 + k0);
    v2f b0 = *(const v2f*)(bp0 + k0);
    v2f b1 = *(const v2f*)(bp1 + k0);
    v2f b2 = *(const v2f*)(bp2 + k0);
    v2f b3 = *(const v2f*)(bp3 + k0);
    acc[0][0] = wmma_f32_k4(a0, b0, acc[0][0]);
    acc[0][1] = wmma_f32_k4(a0, b1, acc[0][1]);
    acc[0][2] = wmma_f32_k4(a0, b2, acc[0][2]);
    acc[0][3] = wmma_f32_k4(a0, b3, acc[0][3]);
    acc[1][0] = wmma_f32_k4(a1, b0, acc[1][0]);
    acc[1][1] = wmma_f32_k4(a1, b1, acc[1][1]);
    acc[1][2] = wmma_f32_k4(a1, b2, acc[1][2]);
    acc[1][3] = wmma_f32_k4(a1, b3, acc[1][3]);
  }

  // C/D layout: c[r] at lane L = C[r + 8*(L>>4)][L&15] within the tile
  const int rbase = (lane >> 4) << 3;
#pragma unroll
  for (int i = 0; i < 2; ++i)
#pragma unroll
    for (int j = 0; j < 4; ++j) {
      float* cp = Cout + (size_t)((tr0 + i) * 16 + rbase) * L_SEQ
                       + (tc0 + j) * 16 + lm;
#pragma unroll
      for (int r = 0; r < 8; ++r) cp[(size_t)r * L_SEQ] = acc[i][j][r];
    }
}

// ---------------------------------------------------------------------------
// Phase 2: per (b,h): sampled-score stats -> M, stable descending ranks
// (exact top_k tie-break: higher value first, lower index wins ties),
// plus the V column means (context term).
// ---------------------------------------------------------------------------
__global__ __launch_bounds__(256) void pa_phase2(
    const float* __restrict__ S, const long long* __restrict__ idx,
    const float* __restrict__ V, int* __restrict__ Mtop,
    float* __restrict__ vmean)
{
  const int bh = blockIdx.x;
  const int b  = bh >> 3;
  const int h  = bh & (H_HEADS - 1);
  const int q  = threadIdx.x;

  const float*     Srow = S + (size_t)bh * MAT_ELEMS + (size_t)q * L_SEQ;
  const long long* irow = idx + (size_t)q * L_SEQ;

  float mx = -3.4028235e38f;
  float sm = 0.f;
  for (int j = 0; j < L_SEQ; ++j) {
    int   jj = (int)irow[j];
    float v  = Srow[jj];
    mx = fmaxf(mx, v);
    sm += v;
  }
  const float M = mx - sm * (1.0f / (float)L_SEQ);

  __shared__ float Ms[L_SEQ];
  Ms[q] = M;
  __syncthreads();

  int rank = 0;
  for (int j = 0; j < L_SEQ; ++j) {
    float mj = Ms[j];
    rank += (mj > M) || (mj == M && j < q);
  }
  Mtop[bh * L_SEQ + rank] = q;   // M_top[rank] = q  (full permutation)

  // vmean[bh][d] = (1/256) * sum_l V[b,l,h,d]   (thread q handles d = q)
  const float* vcol = V + (size_t)b * L_SEQ * ROWSTR + (size_t)h * D_DIM + q;
  float s = 0.f;
  for (int l = 0; l < L_SEQ; ++l) s += vcol[(size_t)l * ROWSTR];
  vmean[bh * L_SEQ + q] = s * (1.0f / (float)L_SEQ);
}

// ---------------------------------------------------------------------------
// Phase 3: out[b, u, h, v] = vmean[v] + SCALE * sum_k S[M_top[u], k] * WT[v][k]
// Same 2x4-tile WMMA blocking; A rows gathered through M_top; epilogue
// applies scale, adds vmean, and writes directly in (B, L, H, D) layout.
// ---------------------------------------------------------------------------
__global__ __launch_bounds__(256) void pa_phase3(
    const float* __restrict__ S, const float* __restrict__ WT,
    const int* __restrict__ Mtop, const float* __restrict__ vmean,
    float* __restrict__ out)
{
  const int bh = blockIdx.y;
  const int b  = bh >> 3;
  const int h  = bh & (H_HEADS - 1);

  const float* Sbh  = S  + (size_t)bh * MAT_ELEMS;
  const float* WTbh = WT + (size_t)bh * MAT_ELEMS;

  const int wave = threadIdx.x >> 5;
  const int lane = threadIdx.x & 31;
  const int lm   = lane & 15;
  const int kk   = (lane >> 4) << 1;

  const int tr0 = blockIdx.x * 4 + (wave >> 2) * 2;
  const int tc0 = (wave & 3) * 4;

  const int arow0 = Mtop[bh * L_SEQ + tr0 * 16 + lm];
  const int arow1 = Mtop[bh * L_SEQ + (tr0 + 1) * 16 + lm];

  const float* ap0 = Sbh  + (size_t)arow0 * L_SEQ + kk;
  const float* ap1 = Sbh  + (size_t)arow1 * L_SEQ + kk;
  const float* bp0 = WTbh + (size_t)((tc0 + 0) * 16 + lm) * L_SEQ + kk;
  const float* bp1 = WTbh + (size_t)((tc0 + 1) * 16 + lm) * L_SEQ + kk;
  const float* bp2 = WTbh + (size_t)((tc0 + 2) * 16 + lm) * L_SEQ + kk;
  const float* bp3 = WTbh + (size_t)((tc0 + 3) * 16 + lm) * L_SEQ + kk;

  const v8f zero = {0.f, 0.f, 0.f, 0.f, 0.f, 0.f, 0.f, 0.f};
  v8f acc[2][4];
  for (int i = 0; i < 2; ++i)
    for (int j = 0; j < 4; ++j) acc[i][j] = zero;

#pragma unroll 8
  for (int k0 = 0; k0 < L_SEQ; k0 += 4) {
    v2f a0 = *(const v2f*)(ap0 + k0);
    v2f a1 = *(const v2f*)(ap1 + k0);
    v2f b0 = *(const v2f*)(bp0 + k0);
    v2f b1 = *(const v2f*)(bp1 + k0);
    v2f b2 = *(const v2f*)(bp2 + k0);
    v2f b3 = *(const v2f*)(bp3 + k0);
    acc[0][0] = wmma_f32_k4(a0, b0, acc[0][0]);
    acc[0][1] = wmma_f32_k4(a0, b1, acc[0][1]);
    acc[0][2] = wmma_f32_k4(a0, b2, acc[0][2]);
    acc[0][3] = wmma_f32_k4(a0, b3, acc[0][3]);
    acc[1][0] = wmma_f32_k4(a1, b0, acc[1][0]);
    acc[1][1] = wmma_f32_k4(a1, b1, acc[1][1]);
    acc[1][2] = wmma_f32_k4(a1, b2, acc[1][2]);
    acc[1][3] = wmma_f32_k4(a1, b3, acc[1][3]);
  }

  const int rbase = (lane >> 4) << 3;
#pragma unroll
  for (int j = 0; j < 4; ++j) {
    const int v  = (tc0 + j) * 16 + lm;
    const float vm = vmean[bh * L_SEQ + v];
#pragma unroll
    for (int i = 0; i < 2; ++i) {
#pragma unroll
      for (int r = 0; r < 8; ++r) {
        const int u = (tr0 + i) * 16 + rbase + r;
        // out layout (B, L, H, D)
        out[(((size_t)b * L_SEQ + u) * H_HEADS + h) * D_DIM + v]
            = vm + SCALE * acc[i][j][r];
      }
    }
  }
}

// ---------------------------------------------------------------------------
extern "C" void kernel_launch(void* const* d_in, const int* in_sizes, int n_in,
                              void* d_out, int out_size, void* d_ws,
                              size_t ws_size, hipStream_t stream) {
  (void)in_sizes; (void)n_in; (void)out_size; (void)ws_size;

  const float*     Q   = (const float*)d_in[0];
  const float*     K   = (const float*)d_in[1];
  const float*     V   = (const float*)d_in[2];
  // d_in[3] = attn_mask (unused by the reference math)
  const long long* idx = (const long long*)d_in[4];   // int64 index_sample
  float* out = (float*)d_out;

  float* wsS   = (float*)d_ws;
  float* wsWT  = wsS  + (size_t)BH * MAT_ELEMS;
  int*   wsTop = (int*)(wsWT + (size_t)BH * MAT_ELEMS);
  float* wsVm  = (float*)(wsTop + BH * L_SEQ);

  pa_phase1<<<dim3(4, 2 * BH), 256, 0, stream>>>(Q, K, V, wsS, wsWT);
  pa_phase2<<<dim3(BH),        256, 0, stream>>>(wsS, idx, V, wsTop, wsVm);
  pa_phase3<<<dim3(4, BH),     256, 0, stream>>>(wsS, wsWT, wsTop, wsVm, out);
}